// DeformConv_68109591380935
// MI455X (gfx1250) — compile-verified
//
#include <hip/hip_runtime.h>
#include <math.h>

typedef float v2f __attribute__((ext_vector_type(2)));
typedef float v8f __attribute__((ext_vector_type(8)));

#define BATCH 4
#define CHI_C 256
#define CHO_C 256
#define H_C 96
#define W_C 96
#define HW_C (H_C * W_C)          // 9216
#define BHW_C (BATCH * HW_C)      // 36864
#define K_C 9
#define NT 32                     // N-tile (pixels) per workgroup
#define NSTRIDE 48                // LDS row stride (floats); 2*48 % 64 == 32 -> conflict-free K-split

// ---------------- workspace layout (bytes) ----------------
// corner records: u32 [8 planes][9 k][BHW]  (planes 0-3: flat offsets, 4-7: weights as bits)
#define CORNER_ELEMS (8u * 9u * (unsigned)BHW_C)
#define CORNER_BYTES ((size_t)CORNER_ELEMS * 4u)
#define W2_ELEMS ((size_t)CHO_C * 9 * CHI_C)
#define W2_BYTES (W2_ELEMS * 4u)
#define STATS_BYTES (2u * 256u * 4u)

// ---------------- kernel 0: weight repack w_conv[o][c][k] -> w2[o][k][c] ----------------
__global__ __launch_bounds__(256) void reorder_k(const float* __restrict__ wconv,
                                                 float* __restrict__ w2) {
    int g = blockIdx.x * 256 + threadIdx.x;          // < CHO*9*CHI
    int c = g & (CHI_C - 1);
    int k = (g >> 8) % 9;
    int o = g / (9 * CHI_C);
    w2[g] = wconv[((size_t)o * CHI_C + c) * 9 + k];
}

// ---------------- kernel 1: offset conv + bilinear corner precompute ----------------
__global__ __launch_bounds__(256) void offs_k(const float* __restrict__ x,
                                              const float* __restrict__ wofs,
                                              const float* __restrict__ bofs,
                                              unsigned* __restrict__ corner) {
    const int g = blockIdx.x * 256 + threadIdx.x;    // < BHW
    const int b = g / HW_C;
    const int hw = g - b * HW_C;
    const int h = hw / W_C;
    const int w = hw - h * W_C;

    float acc[27];
#pragma unroll
    for (int i = 0; i < 27; ++i) acc[i] = bofs[i];

    const float* xb = x + (size_t)b * CHI_C * HW_C;
    for (int ci = 0; ci < CHI_C; ++ci) {
        const float* xp = xb + ci * HW_C;
        float v[9];
#pragma unroll
        for (int t = 0; t < 9; ++t) {
            int yy = h + t / 3 - 1;
            int xx = w + t % 3 - 1;
            bool ok = (yy >= 0) & (yy < H_C) & (xx >= 0) & (xx < W_C);
            v[t] = ok ? xp[yy * W_C + xx] : 0.0f;
        }
        const float* wp = wofs + (size_t)ci * 9;     // + oc*CHI*9
#pragma unroll 3
        for (int oc = 0; oc < 27; ++oc) {
            const float* wo = wp + (size_t)oc * CHI_C * 9;
            float s = acc[oc];
#pragma unroll
            for (int t = 0; t < 9; ++t) s += v[t] * wo[t];
            acc[oc] = s;
        }
    }

#pragma unroll
    for (int k = 0; k < 9; ++k) {
        float dy = acc[2 * k];
        float dx = acc[2 * k + 1];
        float m = 1.0f / (1.0f + expf(-acc[18 + k]));
        float py = (float)(h + k / 3 - 1) + dy;
        float px = (float)(w + k % 3 - 1) + dx;
        float y0f = floorf(py), x0f = floorf(px);
        float ly = py - y0f, lx = px - x0f;
        int y0 = (int)y0f, x0 = (int)x0f;
        int ys[4] = {y0, y0, y0 + 1, y0 + 1};
        int xs[4] = {x0, x0 + 1, x0, x0 + 1};
        float wgt[4] = {(1.0f - ly) * (1.0f - lx), (1.0f - ly) * lx,
                        ly * (1.0f - lx),          ly * lx};
#pragma unroll
        for (int j = 0; j < 4; ++j) {
            bool ok = (ys[j] >= 0) & (ys[j] < H_C) & (xs[j] >= 0) & (xs[j] < W_C);
            int yc = min(max(ys[j], 0), H_C - 1);
            int xc = min(max(xs[j], 0), W_C - 1);
            corner[(size_t)(j * 9 + k) * BHW_C + g] = (unsigned)(yc * W_C + xc);
            corner[(size_t)((j + 4) * 9 + k) * BHW_C + g] =
                __float_as_uint(ok ? wgt[j] * m : 0.0f);
        }
    }
}

// ---------------- kernel 2: fused gather + implicit GEMM (WMMA f32 16x16x4) ----------------
__global__ __launch_bounds__(256) void gemm_k(const float* __restrict__ x,
                                              const float* __restrict__ w2,
                                              const float* __restrict__ bconv,
                                              const unsigned* __restrict__ corner,
                                              float* __restrict__ out) {
    __shared__ float cols[CHI_C * NSTRIDE];          // 48 KB: [K=c][N=n] tile
    __shared__ unsigned cdat[8][NT];                 // corner records for this k

    const int tid = threadIdx.x;
    const int lane = tid & 31;
    const int wave = tid >> 5;                       // 8 waves
    const int n0 = blockIdx.x * NT;
    const int b = n0 / HW_C;                         // HW%NT==0: tile never straddles batch
    const int hw0 = n0 - b * HW_C;
    const float* xb = x + (size_t)b * CHI_C * HW_C;

    v8f acc00 = {0, 0, 0, 0, 0, 0, 0, 0};
    v8f acc01 = {0, 0, 0, 0, 0, 0, 0, 0};
    v8f acc10 = {0, 0, 0, 0, 0, 0, 0, 0};
    v8f acc11 = {0, 0, 0, 0, 0, 0, 0, 0};

    const int row = lane & 15;                       // M row (A) / N col (B, C/D)
    const int kh = lane >> 4;                        // K half select
    const int oR0 = wave * 32 + row;                 // M-tile 0 row
    const int oR1 = wave * 32 + 16 + row;            // M-tile 1 row

    for (int k = 0; k < 9; ++k) {
        __syncthreads();                             // prev-k LDS reads complete
        {
            int j = tid >> 5, i = tid & 31;          // 8*32 == 256 records
            cdat[j][i] = corner[(size_t)(j * 9 + k) * BHW_C + n0 + i];
        }
        __syncthreads();
        {
            const int nloc = tid & 31;
            const int cgrp = tid >> 5;
            const unsigned o0 = cdat[0][nloc], o1 = cdat[1][nloc];
            const unsigned o2 = cdat[2][nloc], o3 = cdat[3][nloc];
            const float f0 = __uint_as_float(cdat[4][nloc]);
            const float f1 = __uint_as_float(cdat[5][nloc]);
            const float f2 = __uint_as_float(cdat[6][nloc]);
            const float f3 = __uint_as_float(cdat[7][nloc]);
            for (int cc = 0; cc < 32; ++cc) {
                const int c = cgrp * 32 + cc;
                const float* xc = xb + (size_t)c * HW_C;
                cols[c * NSTRIDE + nloc] =
                    f0 * xc[o0] + f1 * xc[o1] + f2 * xc[o2] + f3 * xc[o3];
            }
        }
        __syncthreads();

        const float* a0p = w2 + ((size_t)oR0 * 9 + k) * CHI_C + 2 * kh;
        const float* a1p = w2 + ((size_t)oR1 * 9 + k) * CHI_C + 2 * kh;
#pragma unroll 4
        for (int c = 0; c < CHI_C; c += 4) {
            v2f a0 = *(const v2f*)(a0p + c);
            v2f a1 = *(const v2f*)(a1p + c);
            const float* bp = &cols[(c + 2 * kh) * NSTRIDE + row];
            v2f b0 = {bp[0], bp[NSTRIDE]};
            v2f b1 = {bp[16], bp[NSTRIDE + 16]};
            acc00 = __builtin_amdgcn_wmma_f32_16x16x4_f32(false, a0, false, b0,
                                                          (short)0, acc00, false, false);
            acc01 = __builtin_amdgcn_wmma_f32_16x16x4_f32(false, a0, false, b1,
                                                          (short)0, acc01, false, false);
            acc10 = __builtin_amdgcn_wmma_f32_16x16x4_f32(false, a1, false, b0,
                                                          (short)0, acc10, false, false);
            acc11 = __builtin_amdgcn_wmma_f32_16x16x4_f32(false, a1, false, b1,
                                                          (short)0, acc11, false, false);
        }
    }

    // Epilogue: + conv bias; C/D layout: VGPR v -> M = v + 8*kh, N = row
#pragma unroll
    for (int v = 0; v < 8; ++v) {
        int oA = wave * 32 + v + 8 * kh;
        int oB = oA + 16;
        float bA = bconv[oA], bB = bconv[oB];
        size_t baseA = ((size_t)b * CHO_C + oA) * HW_C + hw0 + row;
        size_t baseB = ((size_t)b * CHO_C + oB) * HW_C + hw0 + row;
        out[baseA] = acc00[v] + bA;
        out[baseA + 16] = acc01[v] + bA;
        out[baseB] = acc10[v] + bB;
        out[baseB + 16] = acc11[v] + bB;
    }
}

// ---------------- kernel 3: per-channel mean / rstd ----------------
__global__ __launch_bounds__(256) void stats_k(const float* __restrict__ out,
                                               float* __restrict__ stats) {
    __shared__ float s1[256], s2[256];
    const int o = blockIdx.x;
    const int tid = threadIdx.x;
    float sum = 0.0f, sq = 0.0f;
    for (int idx = tid; idx < BHW_C; idx += 256) {
        int b = idx / HW_C;
        int hw = idx - b * HW_C;
        float v = out[((size_t)b * CHO_C + o) * HW_C + hw];
        sum += v;
        sq += v * v;
    }
    s1[tid] = sum;
    s2[tid] = sq;
    __syncthreads();
    for (int s = 128; s > 0; s >>= 1) {
        if (tid < s) {
            s1[tid] += s1[tid + s];
            s2[tid] += s2[tid + s];
        }
        __syncthreads();
    }
    if (tid == 0) {
        float mu = s1[0] * (1.0f / (float)BHW_C);
        float var = s2[0] * (1.0f / (float)BHW_C) - mu * mu;
        stats[o] = mu;
        stats[256 + o] = rsqrtf(var + 1e-5f);
    }
}

// ---------------- kernel 4: BN + ReLU in place ----------------
__global__ __launch_bounds__(256) void bn_k(float* __restrict__ out,
                                            const float* __restrict__ stats,
                                            const float* __restrict__ gamma,
                                            const float* __restrict__ beta) {
    size_t g = (size_t)blockIdx.x * 256 + threadIdx.x;
    int o = (int)((g / HW_C) % CHO_C);
    float v = out[g];
    float y = gamma[o] * (v - stats[o]) * stats[256 + o] + beta[o];
    out[g] = fmaxf(y, 0.0f);
}

extern "C" void kernel_launch(void* const* d_in, const int* in_sizes, int n_in,
                              void* d_out, int out_size, void* d_ws, size_t ws_size,
                              hipStream_t stream) {
    (void)in_sizes; (void)n_in; (void)out_size; (void)ws_size;
    const float* x        = (const float*)d_in[0];
    const float* w_offset = (const float*)d_in[1];
    const float* b_offset = (const float*)d_in[2];
    const float* w_conv   = (const float*)d_in[3];
    const float* b_conv   = (const float*)d_in[4];
    const float* gamma    = (const float*)d_in[5];
    const float* beta     = (const float*)d_in[6];
    float* out = (float*)d_out;

    char* ws = (char*)d_ws;
    unsigned* corner = (unsigned*)ws;
    float* w2    = (float*)(ws + CORNER_BYTES);
    float* stats = (float*)(ws + CORNER_BYTES + W2_BYTES);

    reorder_k<<<(int)(W2_ELEMS / 256), 256, 0, stream>>>(w_conv, w2);
    offs_k<<<BHW_C / 256, 256, 0, stream>>>(x, w_offset, b_offset, corner);
    gemm_k<<<BHW_C / NT, 256, 0, stream>>>(x, w2, b_conv, corner, out);
    stats_k<<<CHO_C, 256, 0, stream>>>(out, stats);
    bn_k<<<(BATCH * CHO_C * HW_C) / 256, 256, 0, stream>>>(out, stats, gamma, beta);
}